// MultiAttention_49349174231398
// MI455X (gfx1250) — compile-verified
//
#include <hip/hip_runtime.h>

// ---- problem constants ------------------------------------------------------
#define DD   256
#define HH   60
#define WW2  120
#define BB   8
#define HW   7200          // 60*120
// scales: (5,10) -> 12x12=144 windows, (3,6) -> 20x20=400, (2,4) -> 30x30=900

typedef __attribute__((ext_vector_type(2))) float v2f;
typedef __attribute__((ext_vector_type(8))) float v8f;

__device__ __forceinline__ v8f wmma_f32(v2f a, v2f b, v8f c) {
  // D = A(16x4 f32) x B(4x16 f32) + C(16x16 f32), exact fp32 on tensor path
  return __builtin_amdgcn_wmma_f32_16x16x4_f32(false, a, false, b, (short)0, c,
                                               false, false);
}

// ---- K0: transpose the two 256x256 weight matrices --------------------------
__global__ void transpose256(const float* __restrict__ in0,
                             const float* __restrict__ in1,
                             float* __restrict__ out0,
                             float* __restrict__ out1) {
  __shared__ float tile[16][17];
  const float* in = blockIdx.z ? in1 : in0;
  float* out      = blockIdx.z ? out1 : out0;
  int bx = blockIdx.x * 16, by = blockIdx.y * 16;
  int tx = threadIdx.x, ty = threadIdx.y;
  tile[ty][tx] = in[(by + ty) * DD + bx + tx];
  __syncthreads();
  out[(bx + ty) * DD + by + tx] = tile[tx][ty];   // out[r][c] = in[c][r]
}

// ---- K1: 3-scale average pooling, stored channel-major per scale ------------
__global__ void pool_kernel(const float* __restrict__ X,
                            float* __restrict__ P0, float* __restrict__ P1,
                            float* __restrict__ P2) {
  int wid = blockIdx.x % 1444;
  int b   = blockIdx.x / 1444;
  int d   = threadIdx.x;                          // 256 threads == D
  int kh, kw, PH, PW, lw; float* base;
  if (wid < 144)      { lw = wid;       kh = 5; kw = 10; PH = 12; PW = 12; base = P0; }
  else if (wid < 544) { lw = wid - 144; kh = 3; kw = 6;  PH = 20; PW = 20; base = P1; }
  else                { lw = wid - 544; kh = 2; kw = 4;  PH = 30; PW = 30; base = P2; }
  int wy = lw / PW, wx = lw % PW;
  float s = 0.f;
  for (int dy = 0; dy < kh; ++dy)
    for (int dx = 0; dx < kw; ++dx)
      s += X[(((b * HH + wy * kh + dy) * WW2) + wx * kw + dx) * DD + d];
  base[((b * DD + d) * PH + wy) * PW + wx] = s * (1.0f / (float)(kh * kw));
}

// ---- K2: Q = X @ w_q^T + b_q  (M=57600, K=N=256), fp32 WMMA -----------------
__global__ void qgemm(const float* __restrict__ A,     // x rows (57600 x 256)
                      const float* __restrict__ BT,    // wqT: BT[k][j]=w_q[j][k]
                      const float* __restrict__ bias,  // b_q
                      float* __restrict__ Q) {
  __shared__ float As[32][257];   // padded: stride 257 -> conflict-free column reads
  __shared__ float Bs[32][68];
  int t  = threadIdx.x;
  int m0 = blockIdx.x * 32;
  // stage 32x256 A-tile once (float4, fully coalesced)
  for (int it = 0; it < 8; ++it) {
    int fi = it * 256 + t;               // 2048 float4s
    int i = fi >> 6, c4 = fi & 63;
    float4 v = *(const float4*)&A[(size_t)(m0 + i) * DD + c4 * 4];
    As[i][c4 * 4 + 0] = v.x; As[i][c4 * 4 + 1] = v.y;
    As[i][c4 * 4 + 2] = v.z; As[i][c4 * 4 + 3] = v.w;
  }
  int wave = t >> 5, lane = t & 31, h = lane >> 4, ln = lane & 15;
  int rh = (wave >> 2) * 16;            // wave's row-tile offset (0/16)
  int ct = (wave & 3) * 16;             // wave's col-tile offset within 64
  for (int nc = 0; nc < 4; ++nc) {      // 4 x 64 output columns
    v8f acc = {};
    for (int kc = 0; kc < 8; ++kc) {    // K chunks of 32
      __syncthreads();
      for (int it = 0; it < 2; ++it) {  // stage 32x64 B-tile
        int fi = it * 256 + t;
        int rB = fi >> 4, c4 = fi & 15;
        float4 v = *(const float4*)&BT[(size_t)(kc * 32 + rB) * DD + nc * 64 + c4 * 4];
        Bs[rB][c4 * 4 + 0] = v.x; Bs[rB][c4 * 4 + 1] = v.y;
        Bs[rB][c4 * 4 + 2] = v.z; Bs[rB][c4 * 4 + 3] = v.w;
      }
      __syncthreads();
#pragma unroll
      for (int ks = 0; ks < 8; ++ks) {  // 8 x (K=4) WMMA steps
        int kk = ks * 4 + 2 * h;
        v2f a;  a.x = As[rh + ln][kc * 32 + kk]; a.y = As[rh + ln][kc * 32 + kk + 1];
        v2f bv; bv.x = Bs[kk][ct + ln];          bv.y = Bs[kk + 1][ct + ln];
        acc = wmma_f32(a, bv, acc);
      }
    }
#pragma unroll
    for (int r = 0; r < 8; ++r) {       // C/D layout: M = r + 8h, N = ln
      int gn = nc * 64 + ct + ln;
      Q[(size_t)(m0 + rh + r + 8 * h) * DD + gn] = acc[r] + bias[gn];
    }
  }
}

// ---- K3: fused gather-k-GEMM + attn dot -------------------------------------
// k rows follow torch's scrambled view of (B,N,D,H,W) as (B*N, HW, D):
//   row p, col j -> f = p*256+j; dch=f/7200; pix=f%7200 -> pooled cell value.
template <int KH, int KW, int PH, int PW>
__device__ __forceinline__ void fillGather(float (*As)[257],
                                           const float* __restrict__ P,
                                           int b, int p0, int t) {
  for (int it = 0; it < 32; ++it) {          // row=it, col=t (coalesced gather)
    int f   = (p0 + it) * 256 + t;
    int dch = f / HW;
    int r   = f - dch * HW;
    int y   = r / WW2;
    int xx  = r - y * WW2;
    As[it][t] = P[((b * DD + dch) * PH + y / KH) * PW + xx / KW];
  }
}

__global__ void kgemm_attn(const float* __restrict__ P0,
                           const float* __restrict__ P1,
                           const float* __restrict__ P2,
                           const float* __restrict__ BT,   // wkT
                           const float* __restrict__ bk,   // b_k
                           const float* __restrict__ Q,
                           float* __restrict__ attn) {     // (B, HW, 3)
  __shared__ float As[32][257];
  __shared__ float Bs[32][68];
  __shared__ float dotAcc[8][32];
  int t  = threadIdx.x;
  int R0 = blockIdx.x * 32;                 // rows of the (B*3*HW) k matrix
  int bs = R0 / HW;                         // 7200 % 32 == 0: block never straddles
  int b  = bs / 3, n = bs % 3;
  int p0 = R0 - bs * HW;
  if (n == 0)      fillGather<5, 10, 12, 12>(As, P0, b, p0, t);
  else if (n == 1) fillGather<3, 6, 20, 20>(As, P1, b, p0, t);
  else             fillGather<2, 4, 30, 30>(As, P2, b, p0, t);
  dotAcc[t >> 5][t & 31] = 0.f;
  int wave = t >> 5, lane = t & 31, h = lane >> 4, ln = lane & 15;
  int rh = (wave >> 2) * 16, ct = (wave & 3) * 16;
  for (int nc = 0; nc < 4; ++nc) {
    v8f acc = {};
    for (int kc = 0; kc < 8; ++kc) {
      __syncthreads();
      for (int it = 0; it < 2; ++it) {
        int fi = it * 256 + t;
        int rB = fi >> 4, c4 = fi & 15;
        float4 v = *(const float4*)&BT[(size_t)(kc * 32 + rB) * DD + nc * 64 + c4 * 4];
        Bs[rB][c4 * 4 + 0] = v.x; Bs[rB][c4 * 4 + 1] = v.y;
        Bs[rB][c4 * 4 + 2] = v.z; Bs[rB][c4 * 4 + 3] = v.w;
      }
      __syncthreads();
#pragma unroll
      for (int ks = 0; ks < 8; ++ks) {
        int kk = ks * 4 + 2 * h;
        v2f a;  a.x = As[rh + ln][kc * 32 + kk]; a.y = As[rh + ln][kc * 32 + kk + 1];
        v2f bv; bv.x = Bs[kk][ct + ln];          bv.y = Bs[kk + 1][ct + ln];
        acc = wmma_f32(a, bv, acc);
      }
    }
    // fused epilogue: add b_k, multiply by matching Q row, reduce over 16 N-lanes
    int gn = nc * 64 + ct + ln;
    float bkv = bk[gn];
#pragma unroll
    for (int r = 0; r < 8; ++r) {
      int pl = rh + r + 8 * h;                       // local k-row (0..31)
      float v = (acc[r] + bkv) * Q[(size_t)(b * HW + p0 + pl) * DD + gn];
      v += __shfl_xor(v, 1, 16);
      v += __shfl_xor(v, 2, 16);
      v += __shfl_xor(v, 4, 16);
      v += __shfl_xor(v, 8, 16);
      if (ln == 0) dotAcc[wave][pl] += v;            // wave-private column
    }
  }
  __syncthreads();
  if (t < 32) {
    float s = 0.f;
#pragma unroll
    for (int w = 0; w < 8; ++w) s += dotAcc[w][t];
    attn[(size_t)(b * HW + p0 + t) * 3 + n] = s * 0.0625f;  // D^-0.5 = 1/16
  }
}

// ---- K4: per-scale windowed softmax + score + gumbel-sigmoid ----------------
// 30x60 supertile aligns all three window grids exactly.
__global__ void softmax_score(const float* __restrict__ attn,
                              const float* __restrict__ wsc,
                              const float* __restrict__ bsc,
                              const float* __restrict__ ux,
                              const float* __restrict__ ur,
                              float* __restrict__ out) {
  __shared__ float sA[30 * 60 * 3];
  int t  = threadIdx.x;
  int bi = blockIdx.x;
  int b  = bi >> 2;
  int st = bi & 3;
  int row0 = (st >> 1) * 30, col0 = (st & 1) * 60;
  for (int l = t; l < 5400; l += 256) {
    int ly = l / 180, rem = l - ly * 180;
    int lx = rem / 3, i = rem - lx * 3;
    sA[l] = attn[(size_t)(((b * HH + row0 + ly) * WW2) + col0 + lx) * 3 + i];
  }
  __syncthreads();
  // 36 + 100 + 225 = 361 windows, one thread each
  for (int widx = t; widx < 361; widx += 256) {
    int i, kh, kw, gw, lw;
    if (widx < 36)       { i = 0; kh = 5; kw = 10; gw = 6;  lw = widx; }
    else if (widx < 136) { i = 1; kh = 3; kw = 6;  gw = 10; lw = widx - 36; }
    else                 { i = 2; kh = 2; kw = 4;  gw = 15; lw = widx - 136; }
    int wy = lw / gw, wx = lw % gw;
    float mx = -3.4e38f;
    for (int dy = 0; dy < kh; ++dy)
      for (int dx = 0; dx < kw; ++dx)
        mx = fmaxf(mx, sA[(((wy * kh + dy) * 60) + wx * kw + dx) * 3 + i]);
    float sum = 0.f;
    for (int dy = 0; dy < kh; ++dy)
      for (int dx = 0; dx < kw; ++dx)
        sum += __expf(sA[(((wy * kh + dy) * 60) + wx * kw + dx) * 3 + i] - mx);
    float inv = 1.f / sum;
    for (int dy = 0; dy < kh; ++dy)
      for (int dx = 0; dx < kw; ++dx) {
        int idx = (((wy * kh + dy) * 60) + wx * kw + dx) * 3 + i;
        sA[idx] = __expf(sA[idx] - mx) * inv;
      }
  }
  __syncthreads();
  float w0 = wsc[0], w1 = wsc[1], w2 = wsc[2], b0 = bsc[0];
  for (int l = t; l < 1800; l += 256) {
    int ly = l / 60, lx = l - ly * 60;
    float score = sA[l * 3 + 0] * w0 + sA[l * 3 + 1] * w1 + sA[l * 3 + 2] * w2 + b0;
    float s = 1.f / (1.f + expf(-score));
    size_t bp = (size_t)b * HW + (row0 + ly) * WW2 + col0 + lx;
    float lxv = logf(s + 1e-8f);
    float lrv = logf(1.f - s + 1e-8f);
    float nx = -logf(-logf(ux[bp] + 1e-8f) + 1e-8f);
    float nr = -logf(-logf(ur[bp] + 1e-8f) + 1e-8f);
    float Tinv = 1.f / (0.03f + 1e-8f);
    float gx = (lxv + nx) * Tinv;
    float gr = (lrv + nr) * Tinv;
    float m = fmaxf(gx, gr);
    float ex = expf(gx - m), er = expf(gr - m);
    out[bp] = ex / (ex + er);
  }
}

// ---- launcher ---------------------------------------------------------------
extern "C" void kernel_launch(void* const* d_in, const int* in_sizes, int n_in,
                              void* d_out, int out_size, void* d_ws, size_t ws_size,
                              hipStream_t stream) {
  (void)in_sizes; (void)n_in; (void)out_size; (void)ws_size;
  const float* x   = (const float*)d_in[0];
  const float* w_q = (const float*)d_in[1];
  const float* b_q = (const float*)d_in[2];
  const float* w_k = (const float*)d_in[3];
  const float* b_k = (const float*)d_in[4];
  const float* w_s = (const float*)d_in[5];
  const float* b_s = (const float*)d_in[6];
  const float* u_x = (const float*)d_in[7];
  const float* u_r = (const float*)d_in[8];
  float* out = (float*)d_out;

  float* W = (float*)d_ws;
  size_t o = 0;
  float* wqT  = W + o; o += 256 * 256;
  float* wkT  = W + o; o += 256 * 256;
  float* P0   = W + o; o += (size_t)BB * DD * 144;
  float* P1   = W + o; o += (size_t)BB * DD * 400;
  float* P2   = W + o; o += (size_t)BB * DD * 900;
  float* Q    = W + o; o += (size_t)BB * HW * DD;
  float* attn = W + o; o += (size_t)BB * HW * 3;     // ~72 MB total

  transpose256<<<dim3(16, 16, 2), dim3(16, 16), 0, stream>>>(w_q, w_k, wqT, wkT);
  pool_kernel<<<BB * 1444, 256, 0, stream>>>(x, P0, P1, P2);
  qgemm<<<(BB * HW) / 32, 256, 0, stream>>>(x, wqT, b_q, Q);        // 1800 blocks
  kgemm_attn<<<(BB * 3 * HW) / 32, 256, 0, stream>>>(P0, P1, P2, wkT, b_k, Q, attn); // 5400
  softmax_score<<<BB * 4, 256, 0, stream>>>(attn, w_s, b_s, u_x, u_r, out);
}